// Pixel2Voxel_89146341196271
// MI455X (gfx1250) — compile-verified
//
#include <hip/hip_runtime.h>
#include <cstdint>

typedef float v2f __attribute__((ext_vector_type(2)));
typedef float v4f __attribute__((ext_vector_type(4)));
typedef float v8f __attribute__((ext_vector_type(8)));

namespace {
constexpr int W_IMG = 1248;
constexpr int H_IMG = 376;
constexpr int HW    = W_IMG * H_IMG;
constexpr int CR    = 32;   // rgb channels
constexpr int CV    = 16;   // voxel feature channels
constexpr int K3    = 27;   // 3x3x3 kernel offsets
constexpr int CO    = CV + CR;      // 48 output channels
constexpr int BLOCK = 256;          // rows per block (8 wave32)
constexpr int MAXVOX = 12;          // ceil(256/27)+1 = 11 voxels max per block
constexpr int WSB   = 48;           // ws floats per batch
}

// ---------------------------------------------------------------------------
// Setup: per logical batch b (after dummy_batch_list gather) precompute
//   ws[b*48 +  0..15] : Md = rect[bb] @ Trv2c[bb]   (4x4, row major)
//   ws[b*48 + 16..27] : P2[bb]                       (3x4, row major)
//   ws[b*48 + 28..29] : rgb_coor_refine[bb]          (u, v)
//   ws[b*48 + 32..43] : CM = P2[bb] @ Md             (3x4, row major)
// ---------------------------------------------------------------------------
__global__ void setup_mats_kernel(const float* __restrict__ P2,
                                  const float* __restrict__ Trv2c,
                                  const float* __restrict__ rect,
                                  const float* __restrict__ refine,
                                  const int*   __restrict__ dbl,
                                  float* __restrict__ ws, int B) {
  int b = threadIdx.x;
  if (b >= B) return;
  int bb = dbl[b];
  const float* R = rect  + (size_t)bb * 16;
  const float* T = Trv2c + (size_t)bb * 16;
  const float* P = P2    + (size_t)bb * 12;
  float* o = ws + (size_t)b * WSB;
  #pragma unroll
  for (int i = 0; i < 4; ++i) {
    #pragma unroll
    for (int j = 0; j < 4; ++j) {
      float s = R[i * 4 + 0] * T[0 * 4 + j];
      s = fmaf(R[i * 4 + 1], T[1 * 4 + j], s);
      s = fmaf(R[i * 4 + 2], T[2 * 4 + j], s);
      s = fmaf(R[i * 4 + 3], T[3 * 4 + j], s);
      o[i * 4 + j] = s;
    }
  }
  #pragma unroll
  for (int i = 0; i < 12; ++i) o[16 + i] = P[i];
  o[28] = refine[bb * 2 + 0];
  o[29] = refine[bb * 2 + 1];
  // CM = P2 @ Md
  #pragma unroll
  for (int i = 0; i < 3; ++i) {
    #pragma unroll
    for (int j = 0; j < 4; ++j) {
      float s = P[i * 4 + 0] * o[0 * 4 + j];
      s = fmaf(P[i * 4 + 1], o[1 * 4 + j], s);
      s = fmaf(P[i * 4 + 2], o[2 * 4 + j], s);
      s = fmaf(P[i * 4 + 3], o[3 * 4 + j], s);
      o[32 + i * 4 + j] = s;
    }
  }
}

// ---------------------------------------------------------------------------
// Fused projection (WMMA) + bilinear-sample + concat. One thread per row.
// ---------------------------------------------------------------------------
__global__ void __launch_bounds__(BLOCK)
fuse_kernel(const float* __restrict__ vfeat,    // (N, 16)
            const float* __restrict__ rgb,      // (B, 32, H, W)
            const int*   __restrict__ coords,   // (N, 4) : b, z, y, x
            const int*   __restrict__ dbl,      // (B,)
            const float* __restrict__ mats,     // ws, 48 floats per batch
            float* __restrict__ out,            // (M, 48)
            int N, int B) {
  const int M = N * K3;
  const int t = threadIdx.x;
  const int lane = t & 31;
  const int rowBase = blockIdx.x * BLOCK;
  const int n0 = rowBase / K3;

  __shared__ __align__(16) float s_vf[MAXVOX * CV];

  // --- async-stage the voxel features this block will touch (ASYNCcnt) ---
  {
    int lastRow = min(rowBase + BLOCK, M) - 1;
    int nv = lastRow / K3 - n0 + 1;          // <= 11
    if (t < nv * 4) {                        // one b128 (4 floats) per lane
      const float* src = vfeat + ((size_t)n0 * CV + (size_t)t * 4);
      unsigned lds_addr = (unsigned)(uintptr_t)(&s_vf[t * 4]);
      unsigned long long gaddr = (unsigned long long)(uintptr_t)src;
      asm volatile("global_load_async_to_lds_b128 %0, %1, off"
                   :: "v"(lds_addr), "v"(gaddr) : "memory");
    }
    asm volatile("s_wait_asynccnt 0x0" ::: "memory");
  }
  __syncthreads();

  // NOTE: no early return below this point until stores — WMMA and shuffles
  // need EXEC all-ones. Tail lanes clamp the row and skip only the stores.
  const int m  = rowBase + t;
  const int mc = min(m, M - 1);

  // --- per-lane loop-invariant WMMA operand coefficients -----------------
  // A (16x4): row j = 3*jb+ji holds CM_b[ji][:]; only K=0 and K=3 columns
  // are nonzero (middle columns zero => robust to middle-slot layout).
  float a0 = 0.0f, a1 = 0.0f;      // A VGPR0, VGPR1 per documented layout
  {
    int jA = lane & 15;
    if (jA < 3 * B && jA < 12) {
      int jb = jA / 3, ji = jA - 3 * jb;
      const float* cm = mats + (size_t)jb * WSB + 32 + ji * 4;
      if (lane < 16) a0 = cm[0];   // K=0 coefficient (x)
      else           a1 = cm[3];   // K=3 coefficient (1)
    }
  }
  // C accumulator coefficients: row j = v + 8*(lane>=16)
  float cm1[8], cm2[8];
  #pragma unroll
  for (int v = 0; v < 8; ++v) {
    int j = v + ((lane < 16) ? 0 : 8);
    cm1[v] = 0.0f; cm2[v] = 0.0f;
    if (j < 3 * B && j < 12) {
      int jb = j / 3, ji = j - 3 * jb;
      const float* cm = mats + (size_t)jb * WSB + 32 + ji * 4;
      cm1[v] = cm[1];
      cm2[v] = cm[2];
    }
  }

  // --- decode row -> (voxel n, 3x3x3 offset), voxel-center coords --------
  const int n  = mc / K3;
  const int k  = mc - n * K3;
  const int kz = k / 9 - 1;
  const int r3 = k - (k / 9) * 9;
  const int ky = r3 / 3 - 1;
  const int kx = r3 - (r3 / 3) * 3 - 1;

  const int4 vc = *(const int4*)(coords + (size_t)n * 4);
  const int b = vc.x;
  // VOXEL_SIZE = (0.1, 0.05, 0.05) for (z,y,x); COOR_OFFSET = (-3, -40, 0)
  const float cz = ((float)(vc.y + kz) + 0.5f) * 0.10f - 3.0f;
  const float cy = ((float)(vc.z + ky) + 0.5f) * 0.05f - 40.0f;
  const float cx = ((float)(vc.w + kx) + 0.5f) * 0.05f + 0.0f;

  // --- WMMA projection: D[j][n] = CMj0*x_n + CMj1*y_n + CMj2*z_n + CMj3 --
  // Two 16-column chunks cover this wave's 32 rows.
  const int col = lane & 15;
  const float xsA = __shfl(cx, col),      ysA = __shfl(cy, col),      zsA = __shfl(cz, col);
  const float xsB = __shfl(cx, col + 16), ysB = __shfl(cy, col + 16), zsB = __shfl(cz, col + 16);

  v2f Aop = { a0, a1 };
  v2f B1  = { (lane < 16) ? xsA : 0.0f, (lane < 16) ? 0.0f : 1.0f };
  v2f B2  = { (lane < 16) ? xsB : 0.0f, (lane < 16) ? 0.0f : 1.0f };
  v8f C1, C2;
  #pragma unroll
  for (int v = 0; v < 8; ++v) {
    C1[v] = fmaf(cm1[v], ysA, cm2[v] * zsA);
    C2[v] = fmaf(cm1[v], ysB, cm2[v] * zsB);
  }
  v8f D1 = __builtin_amdgcn_wmma_f32_16x16x4_f32(
      false, Aop, false, B1, (short)0, C1, false, false);
  v8f D2 = __builtin_amdgcn_wmma_f32_16x16x4_f32(
      false, Aop, false, B2, (short)0, C2, false, false);

  // --- extract this lane's row: rv[j] = D_half[j][col], j = 0..11 --------
  float rv[12];
  #pragma unroll
  for (int j = 0; j < 12; ++j) {
    const int reg  = (j < 8) ? j : j - 8;
    const int srcl = (j < 8) ? col : col + 16;
    float va = __shfl(D1[reg], srcl);
    float vb = __shfl(D2[reg], srcl);
    rv[j] = (lane < 16) ? va : vb;
  }
  const float img0 = (b == 0) ? rv[0] : (b == 1) ? rv[3] : (b == 2) ? rv[6] : rv[9];
  const float img1 = (b == 0) ? rv[1] : (b == 1) ? rv[4] : (b == 2) ? rv[7] : rv[10];
  const float img2 = (b == 0) ? rv[2] : (b == 1) ? rv[5] : (b == 2) ? rv[8] : rv[11];

  // --- uv, validity, bilinear weights ------------------------------------
  const float* mb = mats + (size_t)b * WSB;
  const float depth = img2;
  const float d = fmaxf(depth, 0.001f);
  const float u = img0 / d + mb[28];
  const float v = img1 / d + mb[29];

  const bool valid = (depth > 0.001f) & (u >= 0.0f) & (u <= (float)(W_IMG - 1)) &
                     (v >= 0.0f) & (v <= (float)(H_IMG - 1));
  const float uc = fminf(fmaxf(u, 0.0f), (float)(W_IMG - 1));
  const float vf = fminf(fmaxf(v, 0.0f), (float)(H_IMG - 1));
  const int x0 = (int)floorf(uc);
  const int y0 = (int)floorf(vf);
  const int x1 = min(x0 + 1, W_IMG - 1);
  const int y1 = min(y0 + 1, H_IMG - 1);
  const float wx = uc - (float)x0;
  const float wy = vf - (float)y0;
  const float vd  = valid ? 1.0f : 0.0f;       // fold validity into weights
  const float w00 = (1.0f - wx) * (1.0f - wy) * vd;
  const float w10 = wx * (1.0f - wy) * vd;
  const float w01 = (1.0f - wx) * wy * vd;
  const float w11 = wx * wy * vd;

  const int i00 = y0 * W_IMG + x0;
  const int i10 = y0 * W_IMG + x1;
  const int i01 = y1 * W_IMG + x0;
  const int i11 = y1 * W_IMG + x1;

  const int bb = dbl[b];
  const float* rb = rgb + (size_t)bb * CR * HW;

  if (m >= M) return;                          // tail lanes: compute only
  float* orow = out + (size_t)m * CO;

  // --- voxel features (from LDS stage) -> out[0..15], streaming stores ---
  {
    const v4f* vfl = (const v4f*)(s_vf + (n - n0) * CV);
    #pragma unroll
    for (int q = 0; q < 4; ++q)
      __builtin_nontemporal_store(vfl[q], (v4f*)orow + q);
  }

  // --- 32 channels of bilinear sampling -> out[16..47] -------------------
  #pragma unroll
  for (int c = 0; c < CR; c += 4) {
    v4f acc;
    // prefetch the next channel group's neighborhood (1.9MB plane stride)
    if (c + 4 < CR)
      __builtin_prefetch(rb + (size_t)(c + 4) * HW + i00, 0, 1);
    #pragma unroll
    for (int cc = 0; cc < 4; ++cc) {
      const float* p = rb + (size_t)(c + cc) * HW;
      float s = w00 * p[i00];
      s = fmaf(w10, p[i10], s);
      s = fmaf(w01, p[i01], s);
      s = fmaf(w11, p[i11], s);
      acc[cc] = s;
    }
    __builtin_nontemporal_store(acc, (v4f*)(orow + CV + c));
  }
}

extern "C" void kernel_launch(void* const* d_in, const int* in_sizes, int n_in,
                              void* d_out, int out_size, void* d_ws, size_t ws_size,
                              hipStream_t stream) {
  const float* vfeat  = (const float*)d_in[0];   // (N,16)
  const float* rgb    = (const float*)d_in[1];   // (B,32,H,W)
  const float* P2     = (const float*)d_in[2];   // (B,3,4)
  const float* Trv2c  = (const float*)d_in[3];   // (B,4,4)
  const float* rect   = (const float*)d_in[4];   // (B,4,4)
  const float* refine = (const float*)d_in[5];   // (B,2)
  const int*   coords = (const int*)d_in[6];     // (N,4)
  const int*   dbl    = (const int*)d_in[7];     // (B,)
  float* ws  = (float*)d_ws;
  float* out = (float*)d_out;

  const int N = in_sizes[0] / CV;
  const int B = in_sizes[7];
  const int M = N * K3;

  setup_mats_kernel<<<1, 64, 0, stream>>>(P2, Trv2c, rect, refine, dbl, ws, B);

  const int grid = (M + BLOCK - 1) / BLOCK;
  fuse_kernel<<<grid, BLOCK, 0, stream>>>(vfeat, rgb, coords, dbl, ws, out, N, B);
}